// SO8TRotationGate_65781719106134
// MI455X (gfx1250) — compile-verified
//
#include <hip/hip_runtime.h>

typedef __attribute__((ext_vector_type(2))) float v2f;
typedef __attribute__((ext_vector_type(4))) float v4f;
typedef __attribute__((ext_vector_type(8))) float v8f;

#define DDIM 4096
#define NBLK 512

// ---------------------------------------------------------------------------
// Kernel 1: A = theta - theta^T ; s = max(ceil(log2(||A||_F + eps)), 0) ;
// R = Taylor-12 of A / (2^s + eps).  One thread per (block, row).
// 16 workgroups x 256 threads = 4096 threads = 512 blocks x 8 rows.
// ---------------------------------------------------------------------------
__global__ __launch_bounds__(256) void so8_expm_taylor(const float* __restrict__ theta,
                                                       float* __restrict__ Rm,
                                                       float* __restrict__ Sv) {
  __shared__ float As_lds[32][64];
  const int t = threadIdx.x;
  const int bl = t >> 3;          // block within workgroup (0..31)
  const int r  = t & 7;           // row within 8x8 block
  const int n  = blockIdx.x * 32 + bl;
  const float* th = theta + n * 64;

  float A[8];
  float sq = 0.0f;
#pragma unroll
  for (int j = 0; j < 8; ++j) {
    A[j] = th[r * 8 + j] - th[j * 8 + r];
    sq += A[j] * A[j];
  }
  // Frobenius-norm^2 reduction across the 8 consecutive lanes of this block
#pragma unroll
  for (int m = 1; m < 8; m <<= 1) sq += __shfl_xor(sq, m, 32);

  const float norm  = sqrtf(sq);
  const float s     = fmaxf(ceilf(log2f(norm + 1e-7f)), 0.0f);
  const float scale = exp2f(s);
  const float inv   = 1.0f / (scale + 1e-7f);

  float As[8];
#pragma unroll
  for (int j = 0; j < 8; ++j) {
    As[j] = A[j] * inv;
    As_lds[bl][r * 8 + j] = As[j];
  }
  __syncthreads();

  float Rrow[8], Ap[8];
#pragma unroll
  for (int j = 0; j < 8; ++j) {
    Ap[j]   = As[j];
    Rrow[j] = ((j == r) ? 1.0f : 0.0f) + As[j];   // i = 1 term
  }
  float fact = 1.0f;
#pragma unroll
  for (int i = 2; i <= 12; ++i) {
    fact *= (float)i;
    float np[8];
#pragma unroll
    for (int j = 0; j < 8; ++j) {
      float acc = 0.0f;
#pragma unroll
      for (int k = 0; k < 8; ++k) acc += Ap[k] * As_lds[bl][k * 8 + j];
      np[j] = acc;
    }
#pragma unroll
    for (int j = 0; j < 8; ++j) {
      Ap[j] = np[j];
      Rrow[j] += np[j] / fact;
    }
  }
#pragma unroll
  for (int j = 0; j < 8; ++j) Rm[n * 64 + r * 8 + j] = Rrow[j];
  if (r == 0) Sv[n] = s;
}

// ---------------------------------------------------------------------------
// Kernel 2: square ALL blocks min((int)max_s, 8) times (faithful to ref).
// max_s is derived identically by every thread -> uniform barrier-safe loop.
// ---------------------------------------------------------------------------
__global__ __launch_bounds__(256) void so8_expm_square(float* __restrict__ Rm,
                                                       const float* __restrict__ Sv) {
  __shared__ float R_lds[32][64];
  const int t = threadIdx.x;
  const int bl = t >> 3;
  const int r  = t & 7;
  const int n  = blockIdx.x * 32 + bl;

  float maxs = 0.0f;
  for (int i = 0; i < NBLK; ++i) maxs = fmaxf(maxs, Sv[i]);   // uniform scalar loads
  const int nsq = (int)fminf(maxs, 8.0f);

  float row[8];
#pragma unroll
  for (int j = 0; j < 8; ++j) row[j] = Rm[n * 64 + r * 8 + j];

  for (int k = 0; k < nsq; ++k) {
#pragma unroll
    for (int j = 0; j < 8; ++j) R_lds[bl][r * 8 + j] = row[j];
    __syncthreads();
    float nr[8];
#pragma unroll
    for (int j = 0; j < 8; ++j) {
      float acc = 0.0f;
#pragma unroll
      for (int e = 0; e < 8; ++e) acc += row[e] * R_lds[bl][e * 8 + j];
      nr[j] = acc;
    }
    __syncthreads();
#pragma unroll
    for (int j = 0; j < 8; ++j) row[j] = nr[j];
  }
#pragma unroll
  for (int j = 0; j < 8; ++j) Rm[n * 64 + r * 8 + j] = row[j];
}

// ---------------------------------------------------------------------------
// Kernel 3: y = x * R_blockdiag via v_wmma_f32_16x16x4_f32.
// Per wave: TWO adjacent 16-channel column pairs (32 channels = 4 blocks =
// one full 128B cacheline per row), so each wave fully consumes the x lines
// it touches (8x b64 loads / row-tile) and fully produces the y lines
// (4x b128 stores / row-tile).  D = Y^T = (R^T blockdiag) x X^T:
//  A (16x4 slices): lane o=L&15, half=L>>4: a[kk][j] = Rdiag[4kk+2*half+j][o]
//  B (4x16 slices): lane = row:            b[kk][j] = x[row][cb+4kk+2*half+j]
//  D (16x16):       lane L holds y[row=L&15][cb + 8*half + 0..7] in c[0..7]
// ---------------------------------------------------------------------------
__global__ __launch_bounds__(256) void so8_apply_wmma(const float* __restrict__ x,
                                                      const float* __restrict__ Rm,
                                                      float* __restrict__ y,
                                                      int ntiles) {
  const int lane = threadIdx.x & 31;
  // wave index is uniform across the wave -> make it scalar for a scalar loop
  const int wave = __builtin_amdgcn_readfirstlane(
      blockIdx.x * (blockDim.x >> 5) + (threadIdx.x >> 5));
  const int cg     = wave & 127;      // 32-channel column group 0..127
  const int rstart = wave >> 7;       // row-tile chunk 0..127
  const int o    = lane & 15;
  const int half = lane >> 4;
  const int cb   = cg * 32;
  const int n0   = cg * 4;            // first of 4 blocks in this group
  const int ob   = o >> 3;            // which 8-block the output channel is in

  // A-matrices = R^T block-diagonal for the two 16x16 tiles, constant per wave
  v2f a0[4], a1[4];
#pragma unroll
  for (int kk = 0; kk < 4; ++kk) {
#pragma unroll
    for (int j = 0; j < 2; ++j) {
      const int e = kk * 4 + half * 2 + j;     // input channel within tile
      float v0 = 0.0f, v1 = 0.0f;
      if ((e >> 3) == ob) {
        v0 = Rm[(n0 + ob) * 64     + (e & 7) * 8 + (o & 7)];
        v1 = Rm[(n0 + 2 + ob) * 64 + (e & 7) * 8 + (o & 7)];
      }
      a0[kk][j] = v0;
      a1[kk][j] = v1;
    }
  }

#pragma unroll 2
  for (int rt = rstart; rt < ntiles; rt += 128) {
    const size_t row = (size_t)rt * 16 + o;            // this lane's row (B/D col)
    const float* xp = x + row * DDIM + cb + half * 2;

    v2f b0[4], b1[4];
#pragma unroll
    for (int kk = 0; kk < 4; ++kk) {
      b0[kk] = *(const v2f*)(xp + kk * 4);
      b1[kk] = *(const v2f*)(xp + 16 + kk * 4);
    }

    v8f c0 = {}, c1 = {};
    c0 = __builtin_amdgcn_wmma_f32_16x16x4_f32(false, a0[0], false, b0[0], (short)0, c0, false, false);
    c1 = __builtin_amdgcn_wmma_f32_16x16x4_f32(false, a1[0], false, b1[0], (short)0, c1, false, false);
    c0 = __builtin_amdgcn_wmma_f32_16x16x4_f32(false, a0[1], false, b0[1], (short)0, c0, false, false);
    c1 = __builtin_amdgcn_wmma_f32_16x16x4_f32(false, a1[1], false, b1[1], (short)0, c1, false, false);
    c0 = __builtin_amdgcn_wmma_f32_16x16x4_f32(false, a0[2], false, b0[2], (short)0, c0, false, false);
    c1 = __builtin_amdgcn_wmma_f32_16x16x4_f32(false, a1[2], false, b1[2], (short)0, c1, false, false);
    c0 = __builtin_amdgcn_wmma_f32_16x16x4_f32(false, a0[3], false, b0[3], (short)0, c0, false, false);
    c1 = __builtin_amdgcn_wmma_f32_16x16x4_f32(false, a1[3], false, b1[3], (short)0, c1, false, false);

    float* yp = y + row * DDIM + cb + half * 8;        // 16B aligned
    v4f l0 = {c0[0], c0[1], c0[2], c0[3]};
    v4f h0 = {c0[4], c0[5], c0[6], c0[7]};
    v4f l1 = {c1[0], c1[1], c1[2], c1[3]};
    v4f h1 = {c1[4], c1[5], c1[6], c1[7]};
    *(v4f*)yp        = l0;
    *(v4f*)(yp + 4)  = h0;
    *(v4f*)(yp + 16) = l1;
    *(v4f*)(yp + 20) = h1;
  }
}

extern "C" void kernel_launch(void* const* d_in, const int* in_sizes, int n_in,
                              void* d_out, int out_size, void* d_ws, size_t ws_size,
                              hipStream_t stream) {
  const float* x     = (const float*)d_in[0];   // [2, 8192, 4096] f32
  const float* theta = (const float*)d_in[1];   // [512, 8, 8] f32
  float* out = (float*)d_out;

  float* Rm = (float*)d_ws;            // 512*64 floats
  float* Sv = Rm + NBLK * 64;          // 512 floats

  const int rows   = in_sizes[0] / DDIM;   // 16384
  const int ntiles = rows / 16;            // 1024

  so8_expm_taylor<<<16, 256, 0, stream>>>(theta, Rm, Sv);
  so8_expm_square<<<16, 256, 0, stream>>>(Rm, Sv);
  // 2048 blocks x 8 waves = 16384 waves = 128 column groups x 128 row chunks
  so8_apply_wmma<<<2048, 256, 0, stream>>>(x, Rm, out, ntiles);
}